// CriticGCNN_4604204941416
// MI455X (gfx1250) — compile-verified
//
#include <hip/hip_runtime.h>
#include <hip/hip_bf16.h>

// ---------------------------------------------------------------------------
// CriticGCNN (4-layer single-head GAT + mean-pool + MLP) for gfx1250 / MI455X
//
// * Dense h = x@W: V_WMMA_F32_16X16X4_F32 (fp32-exact matrix pipe), fed from
//   LDS tiles staged by the Tensor Data Mover (tensor_load_to_lds +
//   s_wait_tensorcnt), 4 row-tiles per wave for B-fragment reuse.
//   __launch_bounds__(32,1) gives the wave the full VGPR file (no spills).
// * Sparse softmax-attention: L2-resident f32/u32 atomics (h/out = 12.8MB
//   each, fully inside the 192MB L2 -> edge traffic never reaches HBM).
// ---------------------------------------------------------------------------

typedef __attribute__((ext_vector_type(2))) float v2f;
typedef __attribute__((ext_vector_type(8))) float v8f;
typedef __attribute__((ext_vector_type(4))) unsigned int tdm_u32x4;
typedef __attribute__((ext_vector_type(8))) int          tdm_i32x8;
typedef __attribute__((ext_vector_type(4))) int          tdm_i32x4;

#define CHN 64            // hidden / out channels
#define NPG 5000          // nodes per graph
#define FCH 512           // fc hidden dim
#define NEG_SLOPE 0.2f
#define RT 4              // row tiles (of 16) per wave in the GEMM

// ---------------------------------------------------------------------------
// TDM: DMA a 2D f32 tile (tile0 x tile1 elements, row stride stride0) from
// global memory into LDS at byte offset lds_off.  Rows beyond dim1 read zero
// (D# OOB rule), which pads the ragged last tile for free.
// D# layout per CDNA5 ISA.
// ---------------------------------------------------------------------------
#if __has_builtin(__builtin_amdgcn_tensor_load_to_lds)
#define USE_TDM 1
__device__ __forceinline__ void tdm_load_tile_f32(unsigned lds_off, const float* gptr,
                                                  unsigned dim0, unsigned dim1,
                                                  unsigned stride0, unsigned tile0,
                                                  unsigned tile1) {
  unsigned long long ga = (unsigned long long)(uintptr_t)gptr;
  tdm_u32x4 g0;
  g0[0] = 1u;                                   // count=1, user-mode, no gather
  g0[1] = lds_off;                              // lds_addr (bytes)
  g0[2] = (unsigned)(ga & 0xFFFFFFFFu);         // global_addr[31:0]
  g0[3] = (unsigned)((ga >> 32) & 0x01FFFFFFu)  // global_addr[56:32]
          | (2u << 30);                         // type = 2 ("image")
  tdm_i32x8 g1 = {};
  g1[0] = (int)(2u << 16);                      // data_size=2 (4B), wg_mask=0
  g1[1] = (int)((dim0 & 0xFFFFu) << 16);        // tensor_dim0[15:0]
  g1[2] = (int)(((dim0 >> 16) & 0xFFFFu) | ((dim1 & 0xFFFFu) << 16));
  g1[3] = (int)(((dim1 >> 16) & 0xFFFFu) | ((tile0 & 0xFFFFu) << 16));
  g1[4] = (int)(tile1 & 0xFFFFu);               // tile_dim1; tile_dim2=0
  g1[5] = (int)stride0;                         // tensor_dim0_stride[31:0]
  tdm_i32x4 z4 = {0, 0, 0, 0};
#if __clang_major__ >= 23
  tdm_i32x8 z8 = {};
  __builtin_amdgcn_tensor_load_to_lds(g0, g1, z4, z4, z8, 0);
#else
  __builtin_amdgcn_tensor_load_to_lds(g0, g1, z4, z4, 0);
#endif
}
#else
#define USE_TDM 0
#endif

// ---------------------------------------------------------------------------
// H[m_base .. m_base+63][0..63] = X[.., 0..KIN) @ W[KIN x 64]
// One wave per block; RT=4 row tiles of 16 -> 16 v8f accumulators (128 VGPRs,
// held entirely in registers thanks to __launch_bounds__(32,1)).
// X tile (64xKIN) and W (KINx64) staged in LDS (TDM when available).
// N is a multiple of 16; last block may own fewer tiles (uniform guard, so
// EXEC stays all-ones around every WMMA).
// ---------------------------------------------------------------------------
template <int KIN>
__global__ __launch_bounds__(32, 1) void gat_gemm_wmma(
    const float* __restrict__ X, const float* __restrict__ W,
    float* __restrict__ H, int n_tiles, int Nn) {
  __shared__ __align__(16) float lds_x[RT * 16 * KIN];  // 64 x KIN tile
  __shared__ __align__(16) float lds_w[KIN * CHN];      // KIN x 64 weights

  const int lane = threadIdx.x & 31;
  const int half = lane >> 4;      // 0: K pair {k,k+1}, 1: K pair {k+2,k+3}
  const int l    = lane & 15;
  const int tile_base = blockIdx.x * RT;
  const int m_base = tile_base * 16;

#if USE_TDM
  tdm_load_tile_f32((unsigned)(uintptr_t)lds_x, X + (size_t)m_base * KIN,
                    (unsigned)KIN, (unsigned)(Nn - m_base), (unsigned)KIN,
                    (unsigned)KIN, (unsigned)(RT * 16));
  tdm_load_tile_f32((unsigned)(uintptr_t)lds_w, W,
                    (unsigned)CHN, (unsigned)KIN, (unsigned)CHN,
                    (unsigned)CHN, (unsigned)KIN);
  __builtin_amdgcn_s_wait_tensorcnt(0);  // single wave: no barrier needed
#else
  for (int i = lane; i < RT * 16 * KIN; i += 32)
    lds_x[i] = ((size_t)m_base * KIN + i < (size_t)Nn * KIN)
                   ? X[(size_t)m_base * KIN + i] : 0.f;
  for (int i = lane; i < KIN * CHN; i += 32) lds_w[i] = W[i];
  __syncthreads();
#endif

  v8f acc[RT][4];
#pragma unroll
  for (int rt = 0; rt < RT; ++rt)
#pragma unroll
    for (int ct = 0; ct < 4; ++ct) acc[rt][ct] = v8f{};

  constexpr bool FULLK = (KIN % 4 == 0);

#pragma unroll
  for (int k0 = 0; k0 < KIN; k0 += 4) {
    const int ka = k0 + 2 * half;
    const int kb = ka + 1;
    const bool kv = FULLK || (kb < KIN);
    const int kbs = kv ? kb : (KIN - 1);     // clamped address, select later

    v2f bf[4];
#pragma unroll
    for (int ct = 0; ct < 4; ++ct) {
      const float by = lds_w[kbs * CHN + ct * 16 + l];
      bf[ct].x = lds_w[ka * CHN + ct * 16 + l];
      bf[ct].y = kv ? by : 0.f;
    }
#pragma unroll
    for (int rt = 0; rt < RT; ++rt) {
      const int row = rt * 16 + l;
      const float ay = lds_x[row * KIN + kbs];
      v2f a;
      a.x = lds_x[row * KIN + ka];
      a.y = kv ? ay : 0.f;
#pragma unroll
      for (int ct = 0; ct < 4; ++ct)
        acc[rt][ct] = __builtin_amdgcn_wmma_f32_16x16x4_f32(
            false, a, false, bf[ct], (short)0, acc[rt][ct], false, false);
    }
  }

  // C/D layout: VGPR r, lanes 0-15 -> M=r, lanes 16-31 -> M=r+8; N = ct*16+l
#pragma unroll
  for (int rt = 0; rt < RT; ++rt) {
    if (tile_base + rt < n_tiles) {          // uniform across the wave
#pragma unroll
      for (int ct = 0; ct < 4; ++ct)
#pragma unroll
        for (int r = 0; r < 8; ++r)
          H[(size_t)(m_base + rt * 16 + r + 8 * half) * CHN + ct * 16 + l] =
              acc[rt][ct][r];
    }
  }
}

// ---------------------------------------------------------------------------
__global__ void fill_zero4_kernel(float4* p, int n4) {
  int t = blockIdx.x * blockDim.x + threadIdx.x;
  if (t < n4) p[t] = float4{0.f, 0.f, 0.f, 0.f};
}

// e_s[r] = h[r].a_s ; e_d[r] = h[r].a_d   (one thread per row, b128 loads)
__global__ void row_alpha_kernel(const float* __restrict__ H,
                                 const float* __restrict__ a_s,
                                 const float* __restrict__ a_d,
                                 float* __restrict__ es, float* __restrict__ ed,
                                 int n) {
  int r = blockIdx.x * blockDim.x + threadIdx.x;
  if (r >= n) return;
  const float4* h4 = (const float4*)(H + (size_t)r * CHN);
  const float4* s4 = (const float4*)a_s;
  const float4* d4 = (const float4*)a_d;
  float s = 0.f, d = 0.f;
#pragma unroll
  for (int c = 0; c < CHN / 4; ++c) {
    float4 hv = h4[c], sv = s4[c], dv = d4[c];
    s += hv.x * sv.x + hv.y * sv.y + hv.z * sv.z + hv.w * sv.w;
    d += hv.x * dv.x + hv.y * dv.y + hv.z * dv.z + hv.w * dv.w;
  }
  es[r] = s;
  ed[r] = d;
}

__device__ __forceinline__ unsigned int f32_order_key(float f) {
  unsigned int b = __float_as_uint(f);
  return (b & 0x80000000u) ? ~b : (b | 0x80000000u);
}
__device__ __forceinline__ float f32_order_unkey(unsigned int k) {
  unsigned int b = (k & 0x80000000u) ? (k & 0x7FFFFFFFu) : ~k;
  return __uint_as_float(b);
}

// per-edge leaky-relu logit + segment max over dst (ordered-uint atomicMax)
__global__ void edge_logit_max_kernel(const int* __restrict__ srcA,
                                      const int* __restrict__ dstA,
                                      const float* __restrict__ es,
                                      const float* __restrict__ ed,
                                      float* __restrict__ eo,
                                      unsigned int* __restrict__ mkey,
                                      int E, int N) {
  int i = blockIdx.x * blockDim.x + threadIdx.x;
  if (i >= E + N) return;
  int s, d;
  if (i < E) { s = srcA[i]; d = dstA[i]; } else { s = i - E; d = s; }  // self loop
  float e = es[s] + ed[d];
  e = e > 0.f ? e : NEG_SLOPE * e;
  eo[i] = e;
  atomicMax(&mkey[d], f32_order_key(e));
}

// p = exp(e - m[dst]); segment sum over dst
__global__ void edge_exp_sum_kernel(const int* __restrict__ dstA,
                                    float* __restrict__ pe,
                                    const unsigned int* __restrict__ mkey,
                                    float* __restrict__ ssum,
                                    int E, int N) {
  int i = blockIdx.x * blockDim.x + threadIdx.x;
  if (i >= E + N) return;
  int d = (i < E) ? dstA[i] : (i - E);
  float m = f32_order_unkey(mkey[d]);
  float p = __expf(pe[i] - m);
  pe[i] = p;
  atomicAdd(&ssum[d], p);
}

// out[dst] += (p/s[dst]) * h[src]  -- 16 threads per edge, float4 each
__global__ void edge_scatter_kernel(const int* __restrict__ srcA,
                                    const int* __restrict__ dstA,
                                    const float* __restrict__ pe,
                                    const float* __restrict__ ssum,
                                    const float* __restrict__ H,
                                    float* __restrict__ OUT,
                                    int E, int N) {
  int t = blockIdx.x * blockDim.x + threadIdx.x;
  int tot = (E + N) * 16;
  if (t >= tot) return;
  int i  = t >> 4;
  int c4 = (t & 15) * 4;
  int s, d;
  if (i < E) { s = srcA[i]; d = dstA[i]; } else { s = i - E; d = s; }
  float alpha = pe[i] / ssum[d];
  const float4 hv = *(const float4*)(H + (size_t)s * CHN + c4);
  float* o = OUT + (size_t)d * CHN + c4;
  atomicAdd(o + 0, alpha * hv.x);
  atomicAdd(o + 1, alpha * hv.y);
  atomicAdd(o + 2, alpha * hv.z);
  atomicAdd(o + 3, alpha * hv.w);
}

// out += bias ; optional relu  (flattened [N,64], b128 accesses)
__global__ void bias_act_kernel(float* __restrict__ OUT, const float* __restrict__ b,
                                int n4, int do_relu) {
  int t = blockIdx.x * blockDim.x + threadIdx.x;
  if (t >= n4) return;
  float4 v = ((float4*)OUT)[t];
  float4 bb = *(const float4*)(b + ((t * 4) & (CHN - 1)));
  v.x += bb.x; v.y += bb.y; v.z += bb.z; v.w += bb.w;
  if (do_relu) {
    v.x = fmaxf(v.x, 0.f); v.y = fmaxf(v.y, 0.f);
    v.z = fmaxf(v.z, 0.f); v.w = fmaxf(v.w, 0.f);
  }
  ((float4*)OUT)[t] = v;
}

// pooled[g] = mean over NPG nodes.  one block (256 thr) per graph.
__global__ void pool_kernel(const float* __restrict__ H, float* __restrict__ pooled) {
  __shared__ float acc[256];
  int g = blockIdx.x, t = threadIdx.x;
  int c = t & (CHN - 1), chunk = t >> 6;  // 4 chunks
  float a = 0.f;
  for (int n = chunk; n < NPG; n += 4)
    a += H[((size_t)g * NPG + n) * CHN + c];
  acc[t] = a;
  __syncthreads();
  if (t < CHN)
    pooled[g * CHN + t] =
        (acc[t] + acc[t + 64] + acc[t + 128] + acc[t + 192]) * (1.0f / NPG);
}

// value[g] = relu(pooled@Wv1 + bv1) @ Wv2 + bv2.  one block (512 thr) per graph.
__global__ void mlp_kernel(const float* __restrict__ pooled,
                           const float* __restrict__ Wv1, const float* __restrict__ bv1,
                           const float* __restrict__ Wv2, const float* __restrict__ bv2,
                           float* __restrict__ value) {
  __shared__ float red[FCH];
  int g = blockIdx.x, j = threadIdx.x;
  float acc = bv1[j];
#pragma unroll
  for (int c = 0; c < CHN; ++c)
    acc += pooled[g * CHN + c] * Wv1[c * FCH + j];
  red[j] = fmaxf(acc, 0.f) * Wv2[j];
  __syncthreads();
  for (int s = FCH / 2; s > 0; s >>= 1) {
    if (j < s) red[j] += red[j + s];
    __syncthreads();
  }
  if (j == 0) value[g] = red[0] + bv2[0];
}

// ---------------------------------------------------------------------------
static inline int cdiv(int a, int b) { return (a + b - 1) / b; }

struct LayerBufs {
  const int* srcA; const int* dstA; int E; int N;
  float *es, *ed, *ssum, *pe; unsigned int* mkey;
};

// One GAT layer: Htmp = Xin@W (WMMA) -> attention scatter into Out (+bias/relu)
template <int KIN>
static void run_gat_layer(const float* Xin, const float* W, const float* a_s,
                          const float* a_d, const float* bias, float* Htmp,
                          float* Out, int do_relu, const LayerBufs& L,
                          hipStream_t stream) {
  const int N = L.N, E = L.E, Etot = E + N;
  const int n_tiles = N / 16;
  gat_gemm_wmma<KIN><<<cdiv(n_tiles, RT), 32, 0, stream>>>(Xin, W, Htmp, n_tiles, N);
  fill_zero4_kernel<<<cdiv(N * CHN / 4, 256), 256, 0, stream>>>((float4*)Out, N * CHN / 4);
  fill_zero4_kernel<<<cdiv(2 * N / 4, 256), 256, 0, stream>>>((float4*)L.mkey, 2 * N / 4);
  row_alpha_kernel<<<cdiv(N, 256), 256, 0, stream>>>(Htmp, a_s, a_d, L.es, L.ed, N);
  edge_logit_max_kernel<<<cdiv(Etot, 256), 256, 0, stream>>>(L.srcA, L.dstA, L.es, L.ed,
                                                             L.pe, L.mkey, E, N);
  edge_exp_sum_kernel<<<cdiv(Etot, 256), 256, 0, stream>>>(L.dstA, L.pe, L.mkey,
                                                           L.ssum, E, N);
  edge_scatter_kernel<<<cdiv(Etot * 16, 256), 256, 0, stream>>>(L.srcA, L.dstA, L.pe,
                                                                L.ssum, Htmp, Out, E, N);
  bias_act_kernel<<<cdiv(N * CHN / 4, 256), 256, 0, stream>>>(Out, bias, N * CHN / 4, do_relu);
}

extern "C" void kernel_launch(void* const* d_in, const int* in_sizes, int n_in,
                              void* d_out, int out_size, void* d_ws, size_t ws_size,
                              hipStream_t stream) {
  // inputs per setup_inputs() order
  const float* x   = (const float*)d_in[0];   // [N,3]
  const int*   ei  = (const int*)d_in[1];     // [2,E] (int32: JAX x64 disabled)
  const float* W1  = (const float*)d_in[2];
  const float* as1 = (const float*)d_in[3];
  const float* ad1 = (const float*)d_in[4];
  const float* b1  = (const float*)d_in[5];
  const float* Wh  = (const float*)d_in[6];   // [2,64,64]
  const float* ash = (const float*)d_in[7];   // [2,64]
  const float* adh = (const float*)d_in[8];
  const float* bh  = (const float*)d_in[9];
  const float* W2  = (const float*)d_in[10];
  const float* as2 = (const float*)d_in[11];
  const float* ad2 = (const float*)d_in[12];
  const float* b2  = (const float*)d_in[13];
  const float* Wv1 = (const float*)d_in[14];
  const float* bv1 = (const float*)d_in[15];
  const float* Wv2 = (const float*)d_in[16];
  const float* bv2 = (const float*)d_in[17];

  const int N = in_sizes[0] / 3;
  const int E = in_sizes[1] / 2;
  const int NG = N / NPG;

  // workspace layout (floats)
  float* hA = (float*)d_ws;                 // [N,64] temp (pre-attention h)
  float* hB = hA + (size_t)N * CHN;         // [N,64] layer output ping-pong
  float* es = hB + (size_t)N * CHN;         // [N]
  float* ed = es + N;                       // [N]
  unsigned int* mkey = (unsigned int*)(ed + N);  // [N]   (mkey,ssum contiguous)
  float* ssum = (float*)(mkey + N);         // [N]
  float* pe = ssum + N;                     // [E+N] edge logits -> probs

  // output layout: [h: N*64][pooled: NG*64][value: NG]
  float* outH   = (float*)d_out;
  float* pooled = outH + (size_t)N * CHN;
  float* value  = pooled + (size_t)NG * CHN;

  LayerBufs L;
  L.srcA = ei; L.dstA = ei + E; L.E = E; L.N = N;
  L.es = es; L.ed = ed; L.ssum = ssum; L.pe = pe; L.mkey = mkey;

  // conv1: [N,3] -> hB, relu
  run_gat_layer<3>(x, W1, as1, ad1, b1, hA, hB, 1, L, stream);
  // hidden convs: hB -> hA (temp) -> hB, relu
  for (int i = 0; i < 2; ++i) {
    run_gat_layer<CHN>(hB, Wh + (size_t)i * CHN * CHN, ash + i * CHN,
                       adh + i * CHN, bh + i * CHN, hA, hB, 1, L, stream);
  }
  // conv2: hB -> d_out h-section, no relu
  run_gat_layer<CHN>(hB, W2, as2, ad2, b2, hA, outH, 0, L, stream);

  // mean pool per graph, then value MLP
  pool_kernel<<<NG, 256, 0, stream>>>(outH, pooled);
  mlp_kernel<<<NG, FCH, 0, stream>>>(pooled, Wv1, bv1, Wv2, bv2, value);
}